// SSM_22505628631461
// MI455X (gfx1250) — compile-verified
//
#include <hip/hip_runtime.h>

#define D_INNER 1024
#define D_STATE 16
#define DT_RANK 64
#define BATCH   4
#define SEQ     1024
#define XP_COLS (DT_RANK + 2 * D_STATE)   /* 96: [0:16)=B, [16:80)=dt_low, [80:96)=C */

typedef float v2f __attribute__((ext_vector_type(2)));
typedef float v8f __attribute__((ext_vector_type(8)));

// ---------------------------------------------------------------------------
// fp32 WMMA GEMM: C[M, NT*16*gridX] = A[M,K] * B[K,N] + bias[N]
// One wave per (16 x NT*16) output stripe; the A fragment is loaded once per
// k-step and reused across NT back-to-back V_WMMA_F32_16X16X4_F32 ops.
// All strides / K are compile-time so B-row loads fold into immediate
// offsets and the k-loop strength-reduces to pointer bumps.
//
// ISA 7.12.2 fragment layouts (wave32):
//   A 16x4 f32 : lane holds A[m][kk + 2*(lane>>4) + v], m = lane&15, v=0,1
//   B 4x16 f32 : lane holds B[kk + 2*(lane>>4) + v][n], n = lane&15, v=0,1
//   C/D 16x16  : vgpr v, lane -> (m = v + 8*(lane>>4), n = lane&15)
// ---------------------------------------------------------------------------
template <int LDA, int LDB, int LDC, int K, int NT>
__global__ __launch_bounds__(32)
void wmma_gemm(const float* __restrict__ A,
               const float* __restrict__ Bm,
               const float* __restrict__ bias,
               float* __restrict__ C)
{
    const int lane  = threadIdx.x & 31;
    const int lh    = lane >> 4;       // half-wave: 0 or 1
    const int ml    = lane & 15;       // row (A) / col (B,C) within tile
    const int tileM = blockIdx.y * 16;
    const int tileN = blockIdx.x * (16 * NT);

    const float* Arow  = A  + (size_t)(tileM + ml) * LDA + 2 * lh;
    const float* Bbase = Bm + (size_t)(2 * lh) * LDB + tileN + ml;

    v8f acc[NT];
#pragma unroll
    for (int nt = 0; nt < NT; ++nt) acc[nt] = {};

#pragma unroll 4
    for (int kk = 0; kk < K; kk += 4) {
        v2f a = *(const v2f*)(Arow + kk);           // A[m][k], A[m][k+1]
#pragma unroll
        for (int nt = 0; nt < NT; ++nt) {
            v2f b;
            b.x = Bbase[(size_t)kk * LDB + nt * 16];            // B[k  ][n]
            b.y = Bbase[(size_t)(kk + 1) * LDB + nt * 16];      // B[k+1][n]
            acc[nt] = __builtin_amdgcn_wmma_f32_16x16x4_f32(
                          /*neg_a=*/false, a, /*neg_b=*/false, b,
                          /*c_mod=*/(short)0, acc[nt],
                          /*reuse_a=*/false, /*reuse_b=*/false);
        }
    }

#pragma unroll
    for (int nt = 0; nt < NT; ++nt) {
        const float bv = bias[tileN + nt * 16 + ml];
        float* Ccol = C + tileN + nt * 16 + ml;
#pragma unroll
        for (int v = 0; v < 8; ++v) {
            const int m = v + 8 * lh;
            Ccol[(size_t)(tileM + m) * LDC] = acc[nt][v] + bv;
        }
    }
}

// ---------------------------------------------------------------------------
// Sequential selective scan. One thread per (b, d); 16 h-states in registers.
//   h[n] <- h[n]*(dt*A[d,n]) + (x*dt)*B_t[n]
//   y     = sum_n h[n]*C_t[n] + D[d]*x
// B_t / C_t rows are read as 8x b128 loads (wave-uniform addresses ->
// broadcast from cache); hs written as 4x b128 per step. Roofline: the 272 MB
// of (ys, hs) stores dominate (~12 us at 23.3 TB/s), so the scan just needs
// enough in-flight stores, which 4096 threads x fire-and-forget b128 gives.
// ---------------------------------------------------------------------------
__global__ __launch_bounds__(256)
void ssm_scan(const float* __restrict__ x,     // (B, S, D)
              const float* __restrict__ xp,    // (B*S, 96)
              const float* __restrict__ dt,    // (B*S, D)
              const float* __restrict__ Amat,  // (D, 16)
              const float* __restrict__ Dvec,  // (D)
              float* __restrict__ ys,          // (S, B, D)
              float* __restrict__ hs)          // (S, B, D, 16)
{
    const int b = blockIdx.x >> 2;                         // 4 blocks per batch
    const int d = ((blockIdx.x & 3) << 8) + threadIdx.x;   // 0..1023

    float h[D_STATE];
    float a[D_STATE];
#pragma unroll
    for (int n = 0; n < D_STATE; ++n) {
        h[n] = 0.0f;
        a[n] = Amat[d * D_STATE + n];
    }
    const float Dd = Dvec[d];

    const float* xb  = x  + (size_t)b * SEQ * D_INNER + d;
    const float* dtb = dt + (size_t)b * SEQ * D_INNER + d;
    const float* xpb = xp + (size_t)b * SEQ * XP_COLS;

    for (int t = 0; t < SEQ; ++t) {
        if (t + 8 < SEQ) {  // stream ahead on the strided x / dt columns
            __builtin_prefetch(xb  + (size_t)(t + 8) * D_INNER, 0, 1);
            __builtin_prefetch(dtb + (size_t)(t + 8) * D_INNER, 0, 1);
        }

        const float xt = xb [(size_t)t * D_INNER];
        const float dv = dtb[(size_t)t * D_INNER];
        const float* row = xpb + (size_t)t * XP_COLS;      // wave-uniform

        // B_t = row[0:16], C_t = row[80:96] as vector loads (16B aligned:
        // row stride 384B, C offset 320B)
        float4 Bv[4], Cv[4];
#pragma unroll
        for (int q = 0; q < 4; ++q) {
            Bv[q] = *(const float4*)(row + 4 * q);
            Cv[q] = *(const float4*)(row + (DT_RANK + D_STATE) + 4 * q);
        }

        const float xdt = xt * dv;
        float y = Dd * xt;

        float* hout = hs + (((size_t)t * BATCH + b) * D_INNER + d) * D_STATE;
#pragma unroll
        for (int q = 0; q < 4; ++q) {
            const float* bq = &Bv[q].x;
            const float* cq = &Cv[q].x;
            float4 hv;
            float* hvp = &hv.x;
#pragma unroll
            for (int j = 0; j < 4; ++j) {
                const int n = 4 * q + j;
                h[n] = h[n] * (dv * a[n]) + xdt * bq[j];
                y   += h[n] * cq[j];
                hvp[j] = h[n];
            }
            *(float4*)(hout + 4 * q) = hv;
        }
        ys[(size_t)t * (BATCH * D_INNER) + (size_t)b * D_INNER + d] = y;
    }
}

// ---------------------------------------------------------------------------
extern "C" void kernel_launch(void* const* d_in, const int* in_sizes, int n_in,
                              void* d_out, int out_size, void* d_ws, size_t ws_size,
                              hipStream_t stream)
{
    const float* x    = (const float*)d_in[0];   // (4, 1024, 1024)
    const float* Wx   = (const float*)d_in[1];   // (1024, 96)
    const float* bx   = (const float*)d_in[2];   // (96)
    const float* Wdt  = (const float*)d_in[3];   // (64, 1024)
    const float* bdt  = (const float*)d_in[4];   // (1024)
    const float* Amat = (const float*)d_in[5];   // (1024, 16)
    const float* Dv   = (const float*)d_in[6];   // (1024)

    float* ys = (float*)d_out;                                   // (S,B,D)
    float* hs = ys + (size_t)SEQ * BATCH * D_INNER;              // (S,B,D,16)

    float* xp    = (float*)d_ws;                                 // (B*S, 96)
    float* dtbuf = xp + (size_t)BATCH * SEQ * XP_COLS;           // (B*S, D)

    // GEMM1: xp = x @ Wx + bx    (M=4096, K=1024, N=96; NT=3 -> 2x256 waves)
    {
        dim3 grid(XP_COLS / (16 * 3), (BATCH * SEQ) / 16);
        wmma_gemm<D_INNER, XP_COLS, XP_COLS, D_INNER, 3>
            <<<grid, 32, 0, stream>>>(x, Wx, bx, xp);
    }
    // GEMM2: dt = xp[:,16:80] @ Wdt + bdt  (M=4096, K=64, N=1024; NT=4)
    {
        dim3 grid(D_INNER / (16 * 4), (BATCH * SEQ) / 16);
        wmma_gemm<XP_COLS, D_INNER, D_INNER, DT_RANK, 4>
            <<<grid, 32, 0, stream>>>(xp + D_STATE, Wdt, bdt, dtbuf);
    }
    // Scan: 16 blocks x 256 threads = one thread per (b, d)
    ssm_scan<<<BATCH * (D_INNER / 256), 256, 0, stream>>>(
        x, xp, dtbuf, Amat, Dv, ys, hs);
}